// Loss_18030272708641
// MI455X (gfx1250) — compile-verified
//
#include <hip/hip_runtime.h>

// ---------------------------------------------------------------------------
// YOLO-v1 loss on MI455X (gfx1250, wave32).
// Memory-bound streaming reduction: ~193 MB in, 4 B out -> ~8.3 us floor at
// 23.3 TB/s. Strategy: 256-cell tiles staged to LDS with coalesced float4
// loads, per-cell math in registers, block reduction via V_WMMA_F32_16X16X4_F32
// (A = partials, B = ones -> row sums), deterministic 2-pass final reduce.
// ---------------------------------------------------------------------------

typedef float v2f __attribute__((ext_vector_type(2)));
typedef float v8f __attribute__((ext_vector_type(8)));

#define CELL_CH 30          // B*5 + C = 2*5 + 20
#define TILE    256         // cells per block (= threads per block)
#define NCH     20          // used channels: 10..29
#define EPSV    1e-6f
#define L_COORD 5.0f
#define L_NOOBJ 0.5f

__device__ __forceinline__ float sgnf(float x) {
    return (x > 0.f) ? 1.f : ((x < 0.f) ? -1.f : 0.f);
}

__global__ __launch_bounds__(TILE)
void yolo_loss_partial(const float* __restrict__ p,
                       const float* __restrict__ a,
                       float* __restrict__ ws,
                       int cells)
{
    __shared__ __align__(16) float sp[TILE * CELL_CH];   // 30720 B
    __shared__ __align__(16) float sa[TILE * CELL_CH];   // 30720 B
    __shared__ float sRed[TILE];

    const int tid        = threadIdx.x;
    const int blockStart = blockIdx.x * TILE;
    const int tileCells  = (cells - blockStart < TILE) ? (cells - blockStart) : TILE;
    const int tileFloats = tileCells * CELL_CH;

    const float* pg = p + (size_t)blockStart * CELL_CH;  // 16B aligned: 256*30*4 = 30720
    const float* ag = a + (size_t)blockStart * CELL_CH;

    // --- cooperative coalesced staging: global b128 -> LDS ---
    const int nVec = tileFloats >> 2;
    const float4* pg4 = (const float4*)pg;
    const float4* ag4 = (const float4*)ag;
    float4* sp4 = (float4*)sp;
    float4* sa4 = (float4*)sa;
    for (int i = tid; i < nVec; i += TILE) {
        sp4[i] = pg4[i];
        sa4[i] = ag4[i];
    }
    for (int i = (nVec << 2) + tid; i < tileFloats; i += TILE) {  // scalar tail (partial tiles)
        sp[i] = pg[i];
        sa[i] = ag[i];
    }
    __syncthreads();

    // --- per-cell loss (one cell per lane; LDS lane stride = 30 floats,
    //     gcd(30,64)=2 -> 32 distinct banks across a wave32: conflict-free) ---
    float acc = 0.f;
    if (tid < tileCells) {
        const float* pc = sp + tid * CELL_CH + 10;  // global ch 10..29 -> local 0..19
        const float* ac = sa + tid * CELL_CH + 10;

        float ptl[2][2], pbr[2][2], parea[2];
        float atl[2][2], abr[2][2], aarea[2];
        float px_[2], py_[2], pwv[2], phv[2], pcf[2];
        float ax_[2], ay_[2], awv[2], ahv[2];
        #pragma unroll
        for (int b = 0; b < 2; ++b) {
            const int o = 10 + 5 * b;  // global ch 20+5b -> local idx
            px_[b] = pc[o];     py_[b] = pc[o + 1];
            pwv[b] = pc[o + 2]; phv[b] = pc[o + 3]; pcf[b] = pc[o + 4];
            ax_[b] = ac[o];     ay_[b] = ac[o + 1];
            awv[b] = ac[o + 2]; ahv[b] = ac[o + 3];
            ptl[b][0] = px_[b] - 0.5f * pwv[b]; ptl[b][1] = py_[b] - 0.5f * phv[b];
            pbr[b][0] = px_[b] + 0.5f * pwv[b]; pbr[b][1] = py_[b] + 0.5f * phv[b];
            parea[b]  = pwv[b] * phv[b];
            atl[b][0] = ax_[b] - 0.5f * awv[b]; atl[b][1] = ay_[b] - 0.5f * ahv[b];
            abr[b][0] = ax_[b] + 0.5f * awv[b]; abr[b][1] = ay_[b] + 0.5f * ahv[b];
            aarea[b]  = awv[b] * ahv[b];
        }

        float max_iou[2];
        #pragma unroll
        for (int pb = 0; pb < 2; ++pb) {
            float m = 0.f;
            #pragma unroll
            for (int ab = 0; ab < 2; ++ab) {
                float sx = fminf(pbr[pb][0], abr[ab][0]) - fmaxf(ptl[pb][0], atl[ab][0]);
                float sy = fminf(pbr[pb][1], abr[ab][1]) - fmaxf(ptl[pb][1], atl[ab][1]);
                sx = fmaxf(sx, 0.f);
                sy = fmaxf(sy, 0.f);
                const float inter = sx * sy;
                const float uni   = parea[pb] + aarea[ab] - inter;
                const float iou   = (uni == 0.f) ? 0.f : inter / uni;
                m = (ab == 0) ? iou : fmaxf(m, iou);
            }
            max_iou[pb] = m;
        }

        const bool obj = ac[14] > 0.f;               // a channel 24 (box0 conf)
        const bool r0  = max_iou[0] >= max_iou[1];   // argmax tie -> first index
        float oij[2];
        oij[0] = (obj && r0)  ? 1.f : 0.f;
        oij[1] = (obj && !r0) ? 1.f : 0.f;

        #pragma unroll
        for (int b = 0; b < 2; ++b) {
            const float o  = oij[b];
            const float dx = ax_[b] - px_[b];
            const float dy = ay_[b] - py_[b];
            const float sw = sgnf(awv[b]) * sqrtf(awv[b] + EPSV)
                           - sgnf(pwv[b]) * sqrtf(pwv[b] + EPSV);
            const float sh = sgnf(ahv[b]) * sqrtf(ahv[b] + EPSV)
                           - sgnf(phv[b]) * sqrtf(phv[b] + EPSV);
            acc += L_COORD * o * (dx * dx + dy * dy + sw * sw + sh * sh);
            const float dc = 1.f - pcf[b];           // (o*o - o*pc)^2 = o*(1-pc)^2
            acc += o * dc * dc;
            acc += L_NOOBJ * (1.f - o) * pcf[b] * pcf[b];  // (noobj*pc)^2
        }

        const float oi = obj ? 1.f : 0.f;            // class loss over ch 10..29
        #pragma unroll
        for (int j = 0; j < NCH; ++j) {
            const float d = pc[j] - ac[j];
            acc += oi * d * d;
        }
    }

    sRed[tid] = acc;
    __syncthreads();

    // --- block reduction via WMMA: 4 chained 16x16x4 f32 ops, B = ones.
    //     A 16x4 lane layout: lane L holds A[L%16][2*(L/16)] , A[L%16][2*(L/16)+1].
    //     D[m][n] = sum_k A[m][k] (+C). Column 0 lives in lanes 0 (m=0..7 over
    //     vgprs) and 16 (m=8..15) -> one shfl_xor(16) finishes the sum. ---
    if (tid < 32) {                                  // full wave32: EXEC all ones
        v8f c = {};
        v2f one; one.x = 1.f; one.y = 1.f;
        #pragma unroll
        for (int chunk = 0; chunk < 4; ++chunk) {
            const int base = chunk * 64 + (tid & 15) * 4 + ((tid >> 4) << 1);
            v2f av; av.x = sRed[base]; av.y = sRed[base + 1];
            c = __builtin_amdgcn_wmma_f32_16x16x4_f32(
                    /*neg_a=*/false, av, /*neg_b=*/false, one,
                    /*c_mod=*/(short)0, c, /*reuse_a=*/false, /*reuse_b=*/false);
        }
        float s = c[0] + c[1] + c[2] + c[3] + c[4] + c[5] + c[6] + c[7];
        s += __shfl_xor(s, 16, 32);
        if (tid == 0) ws[blockIdx.x] = s;
    }
}

__global__ __launch_bounds__(256)
void yolo_loss_final(const float* __restrict__ ws, float* __restrict__ out, int n)
{
    __shared__ double sRed[8];
    double acc = 0.0;
    for (int i = threadIdx.x; i < n; i += 256) acc += (double)ws[i];
    #pragma unroll
    for (int off = 16; off > 0; off >>= 1) acc += __shfl_xor(acc, off, 32);
    if ((threadIdx.x & 31) == 0) sRed[threadIdx.x >> 5] = acc;
    __syncthreads();
    if (threadIdx.x == 0) {
        double t = 0.0;
        #pragma unroll
        for (int w = 0; w < 8; ++w) t += sRed[w];
        out[0] = (float)t;
    }
}

extern "C" void kernel_launch(void* const* d_in, const int* in_sizes, int n_in,
                              void* d_out, int out_size, void* d_ws, size_t ws_size,
                              hipStream_t stream)
{
    (void)n_in; (void)out_size; (void)ws_size;
    const float* p = (const float*)d_in[0];
    const float* a = (const float*)d_in[1];
    float* out = (float*)d_out;
    float* ws  = (float*)d_ws;

    const int cells  = in_sizes[0] / CELL_CH;        // 16384*7*7 = 802816
    const int blocks = (cells + TILE - 1) / TILE;    // 3136 (exact)

    yolo_loss_partial<<<blocks, TILE, 0, stream>>>(p, a, ws, cells);
    yolo_loss_final<<<1, 256, 0, stream>>>(ws, out, blocks);
}